// GridSmoother_7052336300167
// MI455X (gfx1250) — compile-verified
//
#include <hip/hip_runtime.h>
#include <stdint.h>

// GridSmoother: CG solve of (I + Dx^T Wx Dx + Dy^T Wy Dy) x = ae
// B=32, C=16, H=120, W=160. fp32 throughout (CG tol=1e-10 needs it).
// Bandwidth-bound (AI ~0.4 flop/B); ~200MB CG state ~= L2-resident (192MB).
// CDNA5 paths: global_load_async_to_lds_b128 staging (ASYNCcnt) +
// s_wait_asynccnt, and v_wmma_f32_16x16x4_f32 for the wave-level
// dot-product reductions (f32-exact, all-ones B matrix -> row sums).

#define B_ 32
#define C_ 16
#define H_ 120
#define W_ 160
#define TH 8
#define NTILE (H_ / TH)             // 15 tiles per plane
#define NBLK_M (B_ * C_ * NTILE)    // 7680 stencil blocks
#define NBLK_U 2048                 // update-kernel blocks
#define NITER 48                    // >= ~40 iters needed for tol=1e-10 (kappa~9)
#define N_TOT ((size_t)B_ * C_ * H_ * W_)   // 9,830,400 elements

typedef __attribute__((ext_vector_type(2))) float v2f;
typedef __attribute__((ext_vector_type(8))) float v8f;

#if defined(__gfx1250__) && __has_builtin(__builtin_amdgcn_wmma_f32_16x16x4_f32)
#define USE_WMMA_REDUCE 1
#endif

// ---- CDNA5 async global->LDS helpers (ASYNCcnt path) -----------------------

__device__ __forceinline__ uint32_t lds_off(const void* p) {
  // Flat LDS addresses carry the shared aperture in the high 32 bits;
  // low 32 bits are the workgroup-relative LDS byte address (VDST operand).
  return (uint32_t)(uintptr_t)p;
}

__device__ __forceinline__ void async_ld16(uint32_t lds_addr, const float* gaddr) {
  // GLOBAL_LOAD_ASYNC_TO_LDS_B128: per-lane 16B global->LDS DMA, GV mode.
  asm volatile("global_load_async_to_lds_b128 %0, %1, off"
               :: "v"(lds_addr), "v"(gaddr)
               : "memory");
}

__device__ __forceinline__ void wait_async_lds() {
  asm volatile("s_wait_asynccnt 0x0" ::: "memory");
}

// ---- wave-level sum via V_WMMA_F32_16X16X4_F32 -----------------------------
// Pack 32 lane partials as A(16x4) f32: VGPR0 = acc (lanes 0-15 -> K=0 rows
// 0-15; lanes 16-31 -> K=2 rows 0-15), VGPR1 = 0 (K=1,3). B(4x16) = ones.
// D[m][n] = acc_m + acc_{m+16} for every n. Lanes 0-15 hold D rows 0-7 in
// c[0..7], lanes 16-31 hold rows 8-15 -> per-lane sum + shfl_xor(16) = total.
// EXEC must be all ones at the call site (callers invoke at converged points).

__device__ __forceinline__ float wave_sum(float acc) {
#ifdef USE_WMMA_REDUCE
  v2f a;  a[0] = acc; a[1] = 0.f;
  v2f bo; bo[0] = 1.f; bo[1] = 1.f;
  v8f c = {0.f, 0.f, 0.f, 0.f, 0.f, 0.f, 0.f, 0.f};
  c = __builtin_amdgcn_wmma_f32_16x16x4_f32(
      /*neg_a=*/false, a, /*neg_b=*/false, bo,
      /*c_mod=*/(short)0, c, /*reuse_a=*/false, /*reuse_b=*/false);
  float s = ((c[0] + c[1]) + (c[2] + c[3])) + ((c[4] + c[5]) + (c[6] + c[7]));
  return s + __shfl_xor(s, 16, 32);
#else
  float s = acc;
  for (int m = 16; m >= 1; m >>= 1) s += __shfl_xor(s, m, 32);
  return s;
#endif
}

// Combine per-wave sums through a tiny LDS array; thread 0 stores the block
// total in fixed order (deterministic across replays).
__device__ __forceinline__ void block_sum_store(float acc, float* wred,
                                                int tid, int nwaves,
                                                float* dst) {
  const float ws = wave_sum(acc);
  if ((tid & 31) == 0) wred[tid >> 5] = ws;
  __syncthreads();
  if (tid == 0) {
    float t = 0.f;
    for (int i = 0; i < nwaves; ++i) t += wred[i];
    *dst = t;
  }
}

// ---- init: x = b; r = b - A b; p0 = r; partial[blk] = sum(r*r) -------------

__global__ __launch_bounds__(W_) void cg_init_kernel(
    const float* __restrict__ ae, const float* __restrict__ wxwy,
    float* __restrict__ x, float* __restrict__ p0, float* __restrict__ r,
    float* __restrict__ part)
{
  __shared__ float bL[(TH + 2) * W_];
  __shared__ float wxL[TH * W_];
  __shared__ float wyL[(TH + 1) * W_];
  __shared__ float wred[8];

  const int blk = blockIdx.x;
  const int t   = blk % NTILE;
  const int c   = (blk / NTILE) % C_;
  const int b   = blk / (NTILE * C_);
  const int h0  = t * TH;
  const int tid = threadIdx.x;

  const float* wx = wxwy + (size_t)(b * 2 + 0) * H_ * W_;
  const float* wy = wxwy + (size_t)(b * 2 + 1) * H_ * W_;
  const size_t plane = (size_t)(b * C_ + c) * H_ * W_;
  const float* bg = ae + plane;

  const int CHR = W_ / 4;  // 40 x 16B chunks per row
  for (int idx = tid; idx < (TH + 2) * CHR; idx += W_) {
    int row = idx / CHR, col = (idx % CHR) * 4;
    int gh = h0 - 1 + row;
    gh = gh < 0 ? 0 : (gh > H_ - 1 ? H_ - 1 : gh);  // clamped halo (guarded later)
    async_ld16(lds_off(&bL[row * W_ + col]), bg + (size_t)gh * W_ + col);
  }
  for (int idx = tid; idx < TH * CHR; idx += W_) {
    int row = idx / CHR, col = (idx % CHR) * 4;
    async_ld16(lds_off(&wxL[row * W_ + col]), wx + (size_t)(h0 + row) * W_ + col);
  }
  for (int idx = tid; idx < (TH + 1) * CHR; idx += W_) {
    int row = idx / CHR, col = (idx % CHR) * 4;
    int gh = h0 - 1 + row; gh = gh < 0 ? 0 : gh;
    async_ld16(lds_off(&wyL[row * W_ + col]), wy + (size_t)gh * W_ + col);
  }
  wait_async_lds();
  __syncthreads();

  const int w = tid;
  float acc = 0.f;
  for (int i = 0; i < TH; ++i) {
    const int h = h0 + i;
    const float cen = bL[(i + 1) * W_ + w];
    float ax = cen;
    if (w > 0)      ax += wxL[i * W_ + (w - 1)] * (cen - bL[(i + 1) * W_ + (w - 1)]);
    if (w < W_ - 1) ax -= wxL[i * W_ + w]       * (bL[(i + 1) * W_ + (w + 1)] - cen);
    if (h > 0)      ax += wyL[i * W_ + w]       * (cen - bL[i * W_ + w]);
    if (h < H_ - 1) ax -= wyL[(i + 1) * W_ + w] * (bL[(i + 2) * W_ + w] - cen);
    const float rv = cen - ax;                      // r = b - A b
    const size_t g = plane + (size_t)h * W_ + w;
    x[g]  = cen;
    p0[g] = rv;
    r[g]  = rv;
    acc += rv * rv;
  }

  block_sum_store(acc, wred, tid, W_ / 32, &part[blk]);
}

// ---- fused: p_new = r + beta*p_old ; Ap = A p_new ; partial p_new.Ap -------

__global__ __launch_bounds__(W_) void cg_matvec_kernel(
    const float* __restrict__ wxwy,
    const float* __restrict__ pin, float* __restrict__ pout,
    const float* __restrict__ r, float* __restrict__ Ap,
    float* __restrict__ part, const float* __restrict__ scal)
{
  __shared__ float pL[(TH + 2) * W_];
  __shared__ float rL[(TH + 2) * W_];
  __shared__ float wxL[TH * W_];
  __shared__ float wyL[(TH + 1) * W_];
  __shared__ float wred[8];

  const int blk = blockIdx.x;
  const int t   = blk % NTILE;
  const int c   = (blk / NTILE) % C_;
  const int b   = blk / (NTILE * C_);
  const int h0  = t * TH;
  const int tid = threadIdx.x;

  const float beta = scal[3];
  const float* wx = wxwy + (size_t)(b * 2 + 0) * H_ * W_;
  const float* wy = wxwy + (size_t)(b * 2 + 1) * H_ * W_;
  const size_t plane = (size_t)(b * C_ + c) * H_ * W_;
  const float* pg = pin + plane;
  const float* rg = r + plane;

  const int CHR = W_ / 4;
  for (int idx = tid; idx < (TH + 2) * CHR; idx += W_) {
    int row = idx / CHR, col = (idx % CHR) * 4;
    int gh = h0 - 1 + row;
    gh = gh < 0 ? 0 : (gh > H_ - 1 ? H_ - 1 : gh);
    async_ld16(lds_off(&pL[row * W_ + col]), pg + (size_t)gh * W_ + col);
    async_ld16(lds_off(&rL[row * W_ + col]), rg + (size_t)gh * W_ + col);
  }
  for (int idx = tid; idx < TH * CHR; idx += W_) {
    int row = idx / CHR, col = (idx % CHR) * 4;
    async_ld16(lds_off(&wxL[row * W_ + col]), wx + (size_t)(h0 + row) * W_ + col);
  }
  for (int idx = tid; idx < (TH + 1) * CHR; idx += W_) {
    int row = idx / CHR, col = (idx % CHR) * 4;
    int gh = h0 - 1 + row; gh = gh < 0 ? 0 : gh;
    async_ld16(lds_off(&wyL[row * W_ + col]), wy + (size_t)gh * W_ + col);
  }
  wait_async_lds();
  __syncthreads();

  // Phase 1: direction update (elementwise -> halos recomputed locally,
  // which is what makes the fusion race-free with ping-pong p buffers).
  for (int idx = tid; idx < (TH + 2) * W_; idx += W_)
    pL[idx] = rL[idx] + beta * pL[idx];
  __syncthreads();

  // Phase 2: stencil + p.Ap partial
  const int w = tid;
  float acc = 0.f;
  for (int i = 0; i < TH; ++i) {
    const int h = h0 + i;
    const float cen = pL[(i + 1) * W_ + w];
    float ax = cen;
    if (w > 0)      ax += wxL[i * W_ + (w - 1)] * (cen - pL[(i + 1) * W_ + (w - 1)]);
    if (w < W_ - 1) ax -= wxL[i * W_ + w]       * (pL[(i + 1) * W_ + (w + 1)] - cen);
    if (h > 0)      ax += wyL[i * W_ + w]       * (cen - pL[i * W_ + w]);
    if (h < H_ - 1) ax -= wyL[(i + 1) * W_ + w] * (pL[(i + 2) * W_ + w] - cen);
    const size_t g = plane + (size_t)h * W_ + w;
    pout[g] = cen;
    Ap[g]   = ax;
    acc += cen * ax;
  }

  block_sum_store(acc, wred, tid, W_ / 32, &part[blk]);
}

// ---- x += alpha p ; r -= alpha Ap ; partial r.r ----------------------------

__global__ __launch_bounds__(256) void cg_update_kernel(
    float* __restrict__ x, float* __restrict__ r,
    const float* __restrict__ p, const float* __restrict__ Ap,
    float* __restrict__ part, const float* __restrict__ scal)
{
  __shared__ float wred[8];
  const float alpha = scal[2];
  const size_t stride = (size_t)gridDim.x * blockDim.x * 4;
  float acc = 0.f;
  for (size_t i = ((size_t)blockIdx.x * blockDim.x + threadIdx.x) * 4;
       i < N_TOT; i += stride) {
    float4 xv = *(const float4*)(x + i);
    float4 rv = *(const float4*)(r + i);
    float4 pv = *(const float4*)(p + i);
    float4 av = *(const float4*)(Ap + i);
    xv.x += alpha * pv.x; xv.y += alpha * pv.y;
    xv.z += alpha * pv.z; xv.w += alpha * pv.w;
    rv.x -= alpha * av.x; rv.y -= alpha * av.y;
    rv.z -= alpha * av.z; rv.w -= alpha * av.w;
    acc += rv.x * rv.x + rv.y * rv.y + rv.z * rv.z + rv.w * rv.w;
    *(float4*)(x + i) = xv;
    *(float4*)(r + i) = rv;
  }
  block_sum_store(acc, wred, threadIdx.x, 256 / 32, &part[blockIdx.x]);
}

// ---- deterministic scalar reduce + alpha/beta updates ----------------------
// mode 0: rs = sum;            beta = 0            (after init)
// mode 1: pAp = sum;           alpha = rs / pAp    (after matvec)
// mode 2: beta = sum / rs;     rs = sum            (after update)

__global__ __launch_bounds__(256) void cg_reduce_kernel(
    const float* __restrict__ part, int cnt, float* __restrict__ scal, int mode)
{
  __shared__ float wred[8];
  __shared__ float total;
  const int tid = threadIdx.x;
  float acc = 0.f;
  for (int i = tid; i < cnt; i += 256) acc += part[i];
  block_sum_store(acc, wred, tid, 256 / 32, &total);
  __syncthreads();
  if (tid == 0) {
    const float v = total;
    if (mode == 0)      { scal[0] = v; scal[3] = 0.f; }
    else if (mode == 1) { scal[1] = v; scal[2] = scal[0] / v; }
    else                { scal[3] = v / scal[0]; scal[0] = v; }
  }
}

// ---- host-side launch sequence (graph-capture safe: launches only) ---------

extern "C" void kernel_launch(void* const* d_in, const int* in_sizes, int n_in,
                              void* d_out, int out_size, void* d_ws, size_t ws_size,
                              hipStream_t stream) {
  const float* ae   = (const float*)d_in[0];   // [B,C,H,W]
  const float* wxwy = (const float*)d_in[1];   // [B,2,H,W]
  float* x  = (float*)d_out;                   // solution accumulates here

  float* ws  = (float*)d_ws;
  float* p0  = ws;                 // N
  float* p1  = ws + N_TOT;         // N
  float* r   = ws + 2 * N_TOT;     // N
  float* Ap  = ws + 3 * N_TOT;     // N
  float* prt = ws + 4 * N_TOT;     // 8192 block partials
  float* scl = prt + 8192;         // [rs, pAp, alpha, beta]

  cg_init_kernel<<<NBLK_M, W_, 0, stream>>>(ae, wxwy, x, p0, r, prt);
  cg_reduce_kernel<<<1, 256, 0, stream>>>(prt, NBLK_M, scl, 0);

  for (int it = 0; it < NITER; ++it) {
    float* pin  = (it & 1) ? p1 : p0;
    float* pout = (it & 1) ? p0 : p1;
    cg_matvec_kernel<<<NBLK_M, W_, 0, stream>>>(wxwy, pin, pout, r, Ap, prt, scl);
    cg_reduce_kernel<<<1, 256, 0, stream>>>(prt, NBLK_M, scl, 1);
    cg_update_kernel<<<NBLK_U, 256, 0, stream>>>(x, r, pout, Ap, prt, scl);
    if (it + 1 < NITER)
      cg_reduce_kernel<<<1, 256, 0, stream>>>(prt, NBLK_U, scl, 2);
  }
}